// LightGCN_56667798503723
// MI455X (gfx1250) — compile-verified
//
#include <hip/hip_runtime.h>
#include <math.h>

#define NUM_USERS 100000
#define NUM_ITEMS 200000
#define NUM_NODES (NUM_USERS + NUM_ITEMS)
#define EMB_DIM 64
#define NNZ 10000000
#define BATCH 4096
#define WEIGHT_DECAY 1e-4f

typedef __attribute__((ext_vector_type(2))) float v2f;
typedef __attribute__((ext_vector_type(8))) float v8f;

// ---------------- utility kernels ----------------

__global__ void LightGCN_zero_kernel(float* __restrict__ p, long n) {
    long i = (long)blockIdx.x * blockDim.x + threadIdx.x;
    if (i < n) p[i] = 0.0f;
}

// e0 = concat(user_table, item_table); acc = e0; sums[0..1] = 0
__global__ void LightGCN_init_concat(const float* __restrict__ ut,
                                     const float* __restrict__ it,
                                     float* __restrict__ e0,
                                     float* __restrict__ acc,
                                     float* __restrict__ sums) {
    long i = (long)blockIdx.x * blockDim.x + threadIdx.x;
    const long NT = (long)NUM_NODES * EMB_DIM;
    const long UT = (long)NUM_USERS * EMB_DIM;
    if (i < NT) {
        float v = (i < UT) ? ut[i] : it[i - UT];
        e0[i]  = v;
        acc[i] = v;
    }
    if (i < 2) sums[i] = 0.0f;
}

// ---------------- SpMM scatter (memory/atomic bound) ----------------
// One wave32 per edge; lane owns dims [2*lane, 2*lane+1].
// next[r] += v * cur[c];  acc[r] += v * cur[c]  (folds the layer-mean pass in)
__global__ void LightGCN_spmm_scatter(const float* __restrict__ vals,
                                      const int*   __restrict__ rows,
                                      const int*   __restrict__ cols,
                                      const float* __restrict__ cur,
                                      float*       __restrict__ nxt,
                                      float*       __restrict__ acc) {
    long t = (long)blockIdx.x * blockDim.x + threadIdx.x;
    long e = t >> 5;
    int lane = (int)(t & 31);
    if (e >= (long)NNZ) return;

    int   r = rows[e];
    int   c = cols[e];
    float v = vals[e];

    // gfx1250 global_prefetch of a look-ahead source row
    long pe = e + 4096;
    if (pe < (long)NNZ) {
        int pc = cols[pe];
        __builtin_prefetch(cur + (long)pc * EMB_DIM + lane * 2, 0, 3);
    }

    const float2 s = *(const float2*)(cur + (long)c * EMB_DIM + lane * 2);
    float* np = nxt + (long)r * EMB_DIM + lane * 2;
    float* ap = acc + (long)r * EMB_DIM + lane * 2;
    unsafeAtomicAdd(np + 0, v * s.x);
    unsafeAtomicAdd(np + 1, v * s.y);
    unsafeAtomicAdd(ap + 0, v * s.x);
    unsafeAtomicAdd(ap + 1, v * s.y);
}

// ---------------- BPR loss via WMMA f32 16x16x4 ----------------
// One wave32 per tile of 16 batch rows. Two Gram-matrix accumulator chains
// (U·P^T and U·N^T), K=64 folded as 16 x (16x16x4) f32 WMMAs. Diagonal of the
// 16x16 f32 C/D tile per ISA layout: m<8 -> lane m, c[m]; m>=8 -> lane m+16, c[m-8].
__global__ __launch_bounds__(32)
void LightGCN_bpr_wmma(const float* __restrict__ acc,
                       const int* __restrict__ uidx,
                       const int* __restrict__ pidx,
                       const int* __restrict__ nidx,
                       float* __restrict__ sums) {
    int tile = blockIdx.x;           // BATCH/16 tiles
    int lane = threadIdx.x;          // 0..31
    int m    = lane & 15;
    int koff = (lane < 16) ? 0 : 2;  // A/B VGPR0 holds K 0/2 by half-wave, VGPR1 holds K 1/3
    int b    = tile * 16 + m;

    const float* ubase = acc + (long)uidx[b] * EMB_DIM;
    const float* pbase = acc + ((long)NUM_USERS + pidx[b]) * EMB_DIM;
    const float* nbase = acc + ((long)NUM_USERS + nidx[b]) * EMB_DIM;

    v8f cp = {};
    v8f cn = {};
#pragma unroll
    for (int kk = 0; kk < EMB_DIM; kk += 4) {
        float2 af  = *(const float2*)(ubase + kk + koff);
        float2 bpf = *(const float2*)(pbase + kk + koff);
        float2 bnf = *(const float2*)(nbase + kk + koff);
        v2f a;  a.x  = af.x;  a.y  = af.y;
        v2f bp; bp.x = bpf.x; bp.y = bpf.y;
        v2f bn; bn.x = bnf.x; bn.y = bnf.y;
        cp = __builtin_amdgcn_wmma_f32_16x16x4_f32(false, a, false, bp, (short)0, cp, false, false);
        cn = __builtin_amdgcn_wmma_f32_16x16x4_f32(false, a, false, bn, (short)0, cn, false, false);
    }

    // diagonal extraction (select chain; no divergence needed)
    int sel = -1;
    if (lane < 8)        sel = lane;        // row m = lane
    else if (lane >= 24) sel = lane - 24;   // row m = lane - 16
    float dp = 0.0f, dn = 0.0f;
#pragma unroll
    for (int i = 0; i < 8; ++i) {
        dp = (sel == i) ? cp[i] : dp;
        dn = (sel == i) ? cn[i] : dn;
    }

    float contrib = 0.0f;
    if (sel >= 0) {
        // final = acc/4 on both sides -> score scale 1/16
        float x = (dp - dn) * 0.0625f;
        // -log_sigmoid(x) = max(-x,0) + log1p(exp(-|x|))
        contrib = fmaxf(-x, 0.0f) + log1pf(__expf(-fabsf(x)));
    }
#pragma unroll
    for (int off = 16; off >= 1; off >>= 1)
        contrib += __shfl_xor(contrib, off);
    if (lane == 0) unsafeAtomicAdd(&sums[0], contrib);
}

// ---------------- L2 reg over ORIGINAL tables ----------------
// One wave32 per (batch, {user,pos,neg}) row; lane owns a float2.
__global__ void LightGCN_reg_loss(const float* __restrict__ ut,
                                  const float* __restrict__ it,
                                  const int* __restrict__ ui,
                                  const int* __restrict__ pi,
                                  const int* __restrict__ ni,
                                  float* __restrict__ sums) {
    long t = (long)blockIdx.x * blockDim.x + threadIdx.x;
    long w = t >> 5;
    int lane = (int)(t & 31);
    if (w >= (long)BATCH * 3) return;
    int b = (int)(w / 3);
    int typ = (int)(w % 3);
    const float* base;
    if (typ == 0)      base = ut + (long)ui[b] * EMB_DIM;
    else if (typ == 1) base = it + (long)pi[b] * EMB_DIM;
    else               base = it + (long)ni[b] * EMB_DIM;
    float2 s = *(const float2*)(base + lane * 2);
    float v = s.x * s.x + s.y * s.y;
#pragma unroll
    for (int off = 16; off >= 1; off >>= 1)
        v += __shfl_xor(v, off);
    if (lane == 0) unsafeAtomicAdd(&sums[1], v);
}

__global__ void LightGCN_finalize(const float* __restrict__ sums,
                                  float* __restrict__ out) {
    float bpr = sums[0] / (float)BATCH;
    float reg = sums[1] / (float)BATCH;
    out[0] = bpr + WEIGHT_DECAY * reg;
    out[1] = bpr;
    out[2] = reg;
}

// ---------------- launch ----------------

extern "C" void kernel_launch(void* const* d_in, const int* in_sizes, int n_in,
                              void* d_out, int out_size, void* d_ws, size_t ws_size,
                              hipStream_t stream) {
    const float* ut   = (const float*)d_in[0];
    const float* it   = (const float*)d_in[1];
    const float* vals = (const float*)d_in[2];
    const int*   rows = (const int*)d_in[3];
    const int*   cols = (const int*)d_in[4];
    const int*   ui   = (const int*)d_in[5];
    const int*   pi   = (const int*)d_in[6];
    const int*   ni   = (const int*)d_in[7];
    float* out = (float*)d_out;

    const long NT = (long)NUM_NODES * EMB_DIM;
    float* acc  = (float*)d_ws;
    float* e0   = acc + NT;
    float* e1   = e0 + NT;
    float* sums = e1 + NT;   // [0]=bpr sum, [1]=reg sum

    const int tb = 256;
    unsigned int nb_elem = (unsigned int)((NT + tb - 1) / tb);

    LightGCN_init_concat<<<nb_elem, tb, 0, stream>>>(ut, it, e0, acc, sums);

    long scatter_threads = (long)NNZ * 32;
    unsigned int nb_sc = (unsigned int)((scatter_threads + tb - 1) / tb);

    float* cur = e0;
    float* nxt = e1;
    for (int layer = 0; layer < 3; ++layer) {
        LightGCN_zero_kernel<<<nb_elem, tb, 0, stream>>>(nxt, NT);
        LightGCN_spmm_scatter<<<nb_sc, tb, 0, stream>>>(vals, rows, cols, cur, nxt, acc);
        float* tmp = cur; cur = nxt; nxt = tmp;
    }

    LightGCN_bpr_wmma<<<BATCH / 16, 32, 0, stream>>>(acc, ui, pi, ni, sums);

    long reg_threads = (long)BATCH * 3 * 32;
    LightGCN_reg_loss<<<(unsigned int)((reg_threads + tb - 1) / tb), tb, 0, stream>>>(
        ut, it, ui, pi, ni, sums);

    LightGCN_finalize<<<1, 1, 0, stream>>>(sums, out);
}